// Model_47476568490392
// MI455X (gfx1250) — compile-verified
//
#include <hip/hip_runtime.h>

#define BN  16
#define TQN 4000
#define TKN 512
#define EMBN 512
#define ATTN 80
#define NEGT (-0.0005f)

typedef __attribute__((ext_vector_type(2))) float v2f;
typedef __attribute__((ext_vector_type(8))) float v8f;

static __device__ __forceinline__ v8f wmma_f32x4(v2f a, v2f b, v8f c) {
  // V_WMMA_F32_16X16X4_F32 : D(16x16 f32) = A(16x4) * B(4x16) + C
  return __builtin_amdgcn_wmma_f32_16x16x4_f32(
      /*neg_a=*/false, a, /*neg_b=*/false, b,
      /*c_mod=*/(short)0, c, /*reuse_a=*/false, /*reuse_b=*/false);
}

// ---- gfx1250 async global->LDS staging (guarded; falls back to ld/st) ----
#if __has_builtin(__builtin_amdgcn_global_load_async_to_lds_b32)
#define HAVE_ASYNC_LDS 1
typedef __attribute__((address_space(1))) int g_as_int;
typedef __attribute__((address_space(3))) int l_as_int;
#endif

static __device__ __forceinline__ void stage1(const float* g, float* l) {
#ifdef HAVE_ASYNC_LDS
  __builtin_amdgcn_global_load_async_to_lds_b32(
      (g_as_int*)g, (l_as_int*)l, 0, 0);
#else
  *l = *g;  // global_load_b32 + ds_store_b32
#endif
}

static __device__ __forceinline__ void stage_fence() {
#ifdef HAVE_ASYNC_LDS
  asm volatile("s_wait_asynccnt 0x0" ::: "memory");
#endif
}

// x[b,c,t] = emb[ph[b,t], c]   -> out laid out (B, EMB, TK)
__global__ __launch_bounds__(256) void gather_emb(
    const int* __restrict__ ph, const float* __restrict__ emb,
    float* __restrict__ out, int total) {
  int idx = blockIdx.x * 256 + threadIdx.x;
  if (idx >= total) return;
  int t = idx % TKN;
  int rem = idx / TKN;
  int c = rem % EMBN;
  int b = rem / EMBN;
  int p = ph[b * TKN + t];
  out[idx] = emb[(size_t)p * EMBN + c];
}

// Generic 1D conv (cross-correlation, SAME, stride 1) via f32 WMMA.
// in : (B, Cin, T)   w : (Cout, Cin, KW)   out : (B, Cout, T)
// Block = 8 waves sharing one 16-outch slice; weight A-fragments for a
// 64-channel chunk are staged once per block into LDS in fragment order.
template <int T, int KW, bool RELU>
__global__ __launch_bounds__(256) void conv1d_wmma(
    const float* __restrict__ in, const float* __restrict__ w,
    const float* __restrict__ bias, float* __restrict__ out,
    int Cin, int Cout) {
  __shared__ float wlds[1024 * KW];          // 16 groups/chunk * KW * 32 lanes * v2f

  const int wave = threadIdx.x >> 5;
  const int lane = threadIdx.x & 31;
  const int nTT  = T >> 4;
  const int tTile = blockIdx.x * 8 + wave;
  const bool active = (tTile < nTT);         // wave-uniform
  const int t0 = (active ? tTile : 0) << 4;
  const int o0 = blockIdx.y << 4;
  const int b  = blockIdx.z;
  const int m  = lane & 15;                  // M row (outch) / N col (time)
  const int kh = (lane >> 4) << 1;           // K sub-offset {0,2}

  const float* inB = in + (size_t)b * Cin * T;
  const v2f* fr = (const v2f*)wlds;

  v8f acc = {};
  for (int c0 = 0; c0 < Cin; c0 += 64) {
    const int cc  = (Cin - c0 < 64) ? (Cin - c0) : 64;
    const int ccg = cc >> 2;                 // 4-channel K-groups in this chunk

    __syncthreads();                         // LDS reuse: prior reads done
    // ---- cooperative weight staging, fragment order ----
    const int nent = ccg * KW * 32;
    for (int e = threadIdx.x; e < nent; e += 256) {
      const int ls  = e & 31;
      const int grp = e >> 5;
      const int j   = grp / ccg;
      const int cg  = grp - j * ccg;
      const int mm  = ls & 15;
      const int kk  = (ls >> 4) << 1;
      const int c   = c0 + (cg << 2) + kk;
      const float* src = w + ((size_t)(o0 + mm) * Cin + c) * KW + j;
      float* dst = wlds + (((size_t)grp << 5) + ls) * 2;
      stage1(src, dst);
      stage1(src + KW, dst + 1);
    }
    stage_fence();
    __syncthreads();

    if (active) {
      if (c0 + 64 < Cin)                     // pull next input chunk toward L2
        __builtin_prefetch(inB + (size_t)(c0 + 64 + (lane >> 4)) * T + t0 + m, 0, 1);
#pragma unroll
      for (int j = 0; j < KW; ++j) {
        const int t = t0 + m + j - (KW >> 1);
        const bool tin = ((unsigned)t < (unsigned)T);
        const float* bp = inB + (size_t)(c0 + kh) * T + (tin ? t : 0);
        for (int cg = 0; cg < ccg; ++cg) {
          v2f a = fr[(j * ccg + cg) * 32 + lane];   // ds_load_b64
          float b0 = bp[0];                         // offset:0
          float b1 = bp[T];                         // offset:4*T (imm)
          v2f bf;
          bf.x = tin ? b0 : 0.0f;
          bf.y = tin ? b1 : 0.0f;
          acc = wmma_f32x4(a, bf, acc);
          bp += (size_t)4 * T;
        }
      }
    }
  }

  if (active) {
    // D layout: vgpr i, lane L -> o = o0 + i + (L/16)*8 ; t = t0 + (L%16)
    const int halfo = (lane >> 4) << 3;
    float* outB = out + (size_t)b * Cout * T;
    const int tw = t0 + m;
#pragma unroll
    for (int i = 0; i < 8; ++i) {
      const int o = o0 + halfo + i;
      float v = acc[i] + bias[o];
      if (RELU) v = v > 0.0f ? v : 0.0f;
      outB[(size_t)o * T + tw] = v;
    }
  }
}

// out[b,t] = sum_c x[b,c,t]^2     x : (B, C, T)
__global__ __launch_bounds__(256) void sumsq(
    const float* __restrict__ x, float* __restrict__ out,
    int C, int T, int total) {
  int idx = blockIdx.x * 256 + threadIdx.x;
  if (idx >= total) return;
  int t = idx % T;
  int b = idx / T;
  const float* xb = x + (size_t)b * C * T + t;
  float s = 0.0f;
  for (int c = 0; c < C; ++c) {
    float v = xb[(size_t)c * T];
    s += v * v;
  }
  out[idx] = s;
}

// attention[b,t,s] = -T * (q2[b,t] + k2[b,s] - 2 * sum_c q[b,c,t]*k[b,c,s])
// q : (B, ATT, TQ)   k : (B, ATT, TK)   out : (B, TQ, TK)
// Block = 8 waves sharing one 16-row t-tile; q fragments staged in LDS.
__global__ __launch_bounds__(256) void attn_wmma(
    const float* __restrict__ q, const float* __restrict__ k,
    const float* __restrict__ q2, const float* __restrict__ k2,
    float* __restrict__ out) {
  __shared__ float qlds[(ATTN / 4) * 64];    // 20 groups * 32 lanes * v2f = 5KB

  const int wave = threadIdx.x >> 5;
  const int lane = threadIdx.x & 31;
  const int sTile = blockIdx.x * 8 + wave;   // TK/16 = 32 tiles, grid.x = 4
  const int s0 = sTile << 4;
  const int t0 = blockIdx.y << 4;
  const int b  = blockIdx.z;
  const int n  = lane & 15;
  const int kh = (lane >> 4) << 1;

  const float* qb = q + (size_t)b * ATTN * TQN;
  const float* kb = k + (size_t)b * ATTN * TKN;

  // ---- stage q tile (ATT x 16) in fragment order ----
  for (int e = threadIdx.x; e < (ATTN / 4) * 32; e += 256) {
    const int ls  = e & 31;
    const int grp = e >> 5;
    const int mm  = ls & 15;
    const int kk  = (ls >> 4) << 1;
    const int c   = (grp << 2) + kk;
    const float* src = qb + (size_t)c * TQN + t0 + mm;
    float* dst = qlds + (((size_t)grp << 5) + ls) * 2;
    stage1(src, dst);
    stage1(src + TQN, dst + 1);
  }
  stage_fence();
  __syncthreads();

  const v2f* fr = (const v2f*)qlds;
  const float* bp = kb + (size_t)kh * TKN + s0 + n;
  v8f acc = {};
#pragma unroll
  for (int cg = 0; cg < ATTN / 4; ++cg) {    // 20 x v_wmma_f32_16x16x4_f32
    v2f a = fr[(cg << 5) + lane];            // ds_load_b64
    v2f bf;
    bf.x = bp[0];
    bf.y = bp[TKN];
    acc = wmma_f32x4(a, bf, acc);
    bp += (size_t)4 * TKN;
  }

  const int halft = (lane >> 4) << 3;
  const float k2v = k2[(size_t)b * TKN + s0 + n];
  float* outb = out + (size_t)b * TQN * TKN;
#pragma unroll
  for (int i = 0; i < 8; ++i) {
    const int t = t0 + halft + i;
    const float q2v = q2[(size_t)b * TQN + t];
    outb[(size_t)t * TKN + s0 + n] = NEGT * (q2v + k2v - 2.0f * acc[i]);
  }
}

extern "C" void kernel_launch(void* const* d_in, const int* in_sizes, int n_in,
                              void* d_out, int out_size, void* d_ws, size_t ws_size,
                              hipStream_t stream) {
  (void)in_sizes; (void)n_in; (void)out_size; (void)ws_size;
  const float* mels = (const float*)d_in[0];
  const int*   phon = (const int*)d_in[1];
  const float* emb  = (const float*)d_in[2];
  const float* tw1 = (const float*)d_in[3];  const float* tb1 = (const float*)d_in[4];
  const float* tw2 = (const float*)d_in[5];  const float* tb2 = (const float*)d_in[6];
  const float* tw3 = (const float*)d_in[7];  const float* tb3 = (const float*)d_in[8];
  const float* ow1 = (const float*)d_in[9];  const float* ob1 = (const float*)d_in[10];
  const float* ow2 = (const float*)d_in[11]; const float* ob2 = (const float*)d_in[12];
  const float* mw1 = (const float*)d_in[13]; const float* mb1 = (const float*)d_in[14];
  const float* mw2 = (const float*)d_in[15]; const float* mb2 = (const float*)d_in[16];
  const float* mw3 = (const float*)d_in[17]; const float* mb3 = (const float*)d_in[18];

  // ---- workspace layout (floats) ----
  float* ws = (float*)d_ws;
  size_t off = 0;
  float* kbuf = ws + off; off += (size_t)BN * ATTN * TKN;   //   655,360
  float* k2b  = ws + off; off += (size_t)BN * TKN;          //     8,192
  float* qbuf = ws + off; off += (size_t)BN * ATTN * TQN;   // 5,120,000
  float* q2b  = ws + off; off += (size_t)BN * TQN;          //    64,000
  float* pool = ws + off;
  // text phase (runs first):
  float* xa = pool;
  float* xb = pool + (size_t)BN * EMBN * TKN;
  float* yb = pool + 2 * (size_t)BN * EMBN * TKN;           // (B,1024,TK)
  // mel phase (aliases pool after text phase completes):
  float* m1 = pool;                                         // (B,256,TQ)
  float* m2 = pool + (size_t)BN * 256 * TQN;                // (B,128,TQ)

  // ---- text/key path ----
  {
    int total = BN * EMBN * TKN;
    gather_emb<<<(total + 255) / 256, 256, 0, stream>>>(phon, emb, xa, total);
  }
  {
    dim3 g((TKN / 16 + 7) / 8, EMBN / 16, BN);
    conv1d_wmma<TKN, 5, true><<<g, 256, 0, stream>>>(xa, tw1, tb1, xb, EMBN, EMBN);
    conv1d_wmma<TKN, 5, true><<<g, 256, 0, stream>>>(xb, tw2, tb2, xa, EMBN, EMBN);
    conv1d_wmma<TKN, 5, true><<<g, 256, 0, stream>>>(xa, tw3, tb3, xb, EMBN, EMBN);
  }
  {
    dim3 g((TKN / 16 + 7) / 8, (2 * EMBN) / 16, BN);
    conv1d_wmma<TKN, 3, true><<<g, 256, 0, stream>>>(xb, ow1, ob1, yb, EMBN, 2 * EMBN);
  }
  {
    dim3 g((TKN / 16 + 7) / 8, ATTN / 16, BN);
    conv1d_wmma<TKN, 1, false><<<g, 256, 0, stream>>>(yb, ow2, ob2, kbuf, 2 * EMBN, ATTN);
  }
  sumsq<<<(BN * TKN + 255) / 256, 256, 0, stream>>>(kbuf, k2b, ATTN, TKN, BN * TKN);

  // ---- mel/query path ----
  {
    dim3 g((TQN / 16 + 7) / 8, 256 / 16, BN);
    conv1d_wmma<TQN, 3, true><<<g, 256, 0, stream>>>(mels, mw1, mb1, m1, 80, 256);
  }
  {
    dim3 g((TQN / 16 + 7) / 8, 128 / 16, BN);
    conv1d_wmma<TQN, 3, true><<<g, 256, 0, stream>>>(m1, mw2, mb2, m2, 256, 128);
  }
  {
    dim3 g((TQN / 16 + 7) / 8, ATTN / 16, BN);
    conv1d_wmma<TQN, 1, false><<<g, 256, 0, stream>>>(m2, mw3, mb3, qbuf, 128, ATTN);
  }
  sumsq<<<(BN * TQN + 255) / 256, 256, 0, stream>>>(qbuf, q2b, ATTN, TQN, BN * TQN);

  // ---- fused distance attention ----
  {
    dim3 g((TKN / 16) / 8, TQN / 16, BN);
    attn_wmma<<<g, 256, 0, stream>>>(qbuf, kbuf, q2b, k2b, (float*)d_out);
  }
}